// WindowAttention_59674275611081
// MI455X (gfx1250) — compile-verified
//
#include <hip/hip_runtime.h>
#include <math.h>

// ---------------------------------------------------------------------------
// WindowAttention on MI455X (gfx1250, wave32, WMMA + TDM/async-LDS)
//
// Memory-bound problem (0.31 TFLOP fp32 vs ~1.85GB minimal traffic at
// 23.3 TB/s => ~80us memory vs ~1us fp32-WMMA math). Keep fp32 and use
// V_WMMA_F32_16X16X4_F32 for all GEMMs; fuse qkv-GEMM+GELU+attention per
// window so the (B,S,3C) qkv tensor lives only in LDS. Stage the x window
// with the Tensor Data Mover and GEMM tiles with async global->LDS loads.
// ---------------------------------------------------------------------------

#define WS   7
#define SEQ  49      // tokens per window
#define SP   64      // padded sequence (4 x 16 tiles)
#define CH   384     // dim
#define NHD  12      // heads
#define HD   32      // head dim
#define NB   4096    // windows
#define MTOT (NB * SEQ)   // 200704 rows

typedef __attribute__((ext_vector_type(2))) float        v2f;
typedef __attribute__((ext_vector_type(8))) float        v8f;
typedef __attribute__((ext_vector_type(4))) unsigned int v4u;
typedef __attribute__((ext_vector_type(8))) int          v8i;
typedef __attribute__((ext_vector_type(4))) int          v4i;

#if defined(__has_builtin)
#  if __has_builtin(__builtin_amdgcn_tensor_load_to_lds)
#    define HAVE_TDM 1
#  else
#    define HAVE_TDM 0
#  endif
#else
#  define HAVE_TDM 0
#endif

__device__ __forceinline__ float gelu_exact(float v) {
    return 0.5f * v * (1.0f + erff(v * 0.70710678118654752f));
}

// ---------------------------------------------------------------------------
// Kernel 1: per-window fused  qkv = gelu(x@Wqkv+b); attn; out -> (B,S,C)
// 256 threads = 8 waves (wave32). One block per window, loop over 12 heads.
// LDS: x tile 64x384 zero-padded (96KB) + q/k/v 64x32 (24KB) + sc 64x64 (16KB).
// ---------------------------------------------------------------------------
__global__ __launch_bounds__(256) void fused_qkv_attn(
    const float* __restrict__ x,          // (B,S,C)
    const float* __restrict__ w_qkv,      // (C, 3C)
    const float* __restrict__ b_qkv,      // (3C)
    const float* __restrict__ bias_table, // (313, NH)
    float* __restrict__ attn_out)         // (B,S,C)
{
    __shared__ float xs[SP * CH];         // rows 49..63 zeroed
    __shared__ float qs[SP * HD];
    __shared__ float ks[SP * HD];
    __shared__ float vs[SP * HD];
    __shared__ float sc[SP * SP];

    const int b        = blockIdx.x;
    const int tid      = threadIdx.x;
    const int lane     = tid & 31;
    const int wave     = tid >> 5;
    const int laneM    = lane & 15;    // M (and N) index within 16x16 tile
    const int laneHalf = lane >> 4;    // 0: K={0,1}, 1: K={2,3}
    const int kOff     = laneHalf * 2;

    // ---- Stage the x window (49x384 fp32, contiguous) into LDS ------------
    const float* xg = x + (size_t)b * SEQ * CH;
#if HAVE_TDM
    if (wave == 0) {
        // Tensor DMA descriptor: 2D tile 384x49 of 4-byte elements.
        const unsigned lds_addr = (unsigned)(uintptr_t)(void*)xs;
        const unsigned long long ga = (unsigned long long)(uintptr_t)xg;
        v4u g0;
        g0[0] = 1u;                                    // count=1, user mode
        g0[1] = lds_addr;                              // lds_addr
        g0[2] = (unsigned)(ga & 0xffffffffu);          // global_addr[31:0]
        g0[3] = (unsigned)((ga >> 32) & 0x01ffffffu)   // global_addr[56:32]
                | 0x80000000u;                         // type=2 ("image")
        v8i g1;
        g1[0] = 0x00020000;        // workgroup_mask=0, data_size=2 (4B)
        g1[1] = (int)(384u << 16); // tensor_dim0 = 384 (bits 79:48 low half)
        g1[2] = (int)(49u << 16);  // tensor_dim1 = 49  (bits 111:80 low half)
        g1[3] = (int)(384u << 16); // tile_dim0 = 384   (bits 127:112)
        g1[4] = 49;                // tile_dim1 = 49, tile_dim2 = 0
        g1[5] = 384;               // tensor_dim0_stride[31:0] = 384
        g1[6] = 0;                 // stride0 hi, tensor_dim1_stride lo
        g1[7] = 0;
        v4i g2 = {0, 0, 0, 0};
        v4i g3 = {0, 0, 0, 0};
        v8i g4 = {0, 0, 0, 0, 0, 0, 0, 0};   // extra group slot (clang-23 ABI)
        __builtin_amdgcn_tensor_load_to_lds(g0, g1, g2, g3, g4, 0);
        asm volatile("s_wait_tensorcnt 0x0" ::: "memory");
    }
    for (int i = tid; i < (SP - SEQ) * CH; i += 256) xs[SEQ * CH + i] = 0.f;
#else
    for (int i = tid; i < SEQ * CH; i += 256) xs[i] = xg[i];
    for (int i = tid; i < (SP - SEQ) * CH; i += 256) xs[SEQ * CH + i] = 0.f;
#endif
    __syncthreads();

    for (int h = 0; h < NHD; ++h) {
        // ------- Phase 1: q/k/v = gelu(x @ Wqkv[:, mat*C + h*32 + :]) ------
        // 24 wave-tiles: mat(3) x mtile(4) x ntile(2); 3 tiles per wave.
        for (int t = wave * 3; t < wave * 3 + 3; ++t) {
            const int mat  = t >> 3;          // 0=q 1=k 2=v
            const int rem  = t & 7;
            const int mt   = rem >> 1;        // row tile 0..3
            const int nt   = rem & 1;         // col tile 0..1
            const int row0 = mt * 16;
            const int colW = mat * CH + h * HD + nt * 16;  // col in w_qkv

            v8f acc = {};
            const int ar = row0 + laneM;      // padded rows are zero in LDS
            for (int k0 = 0; k0 < CH; k0 += 4) {
                v2f a, bb;
                a.x  = xs[ar * CH + k0 + kOff];
                a.y  = xs[ar * CH + k0 + kOff + 1];
                bb.x = w_qkv[(size_t)(k0 + kOff)     * (3 * CH) + colW + laneM];
                bb.y = w_qkv[(size_t)(k0 + kOff + 1) * (3 * CH) + colW + laneM];
                acc = __builtin_amdgcn_wmma_f32_16x16x4_f32(
                          false, a, false, bb, (short)0, acc, false, false);
            }
            const float bqv = b_qkv[colW + laneM];
            float* dst = (mat == 0) ? qs : (mat == 1) ? ks : vs;
            const int col = nt * 16 + laneM;
            #pragma unroll
            for (int r = 0; r < 8; ++r) {
                const int row = row0 + r + laneHalf * 8;
                dst[row * HD + col] = gelu_exact(acc[r] + bqv);
            }
        }
        __syncthreads();

        // ------- Phase 2: scores = q @ k^T + rel-pos bias  (64x64) ---------
        for (int t = wave * 2; t < wave * 2 + 2; ++t) {
            const int mt = t >> 2, nt = t & 3;
            const int row0 = mt * 16, col0 = nt * 16;
            v8f acc = {};
            #pragma unroll
            for (int k0 = 0; k0 < HD; k0 += 4) {
                v2f a, bb;
                a.x  = qs[(row0 + laneM) * HD + k0 + kOff];
                a.y  = qs[(row0 + laneM) * HD + k0 + kOff + 1];
                bb.x = ks[(col0 + laneM) * HD + k0 + kOff];     // B = k^T
                bb.y = ks[(col0 + laneM) * HD + k0 + kOff + 1];
                acc = __builtin_amdgcn_wmma_f32_16x16x4_f32(
                          false, a, false, bb, (short)0, acc, false, false);
            }
            #pragma unroll
            for (int r = 0; r < 8; ++r) {
                const int row = row0 + r + laneHalf * 8;
                const int col = col0 + laneM;
                float bv = 0.f;
                if (row < SEQ && col < SEQ) {
                    const int idx = 13 * ((col % 7 - row % 7) + (col / 7 - row / 7) + 12);
                    bv = bias_table[idx * NHD + h];
                }
                sc[row * SP + col] = acc[r] + bv;
            }
        }
        __syncthreads();

        // ------- Phase 3: softmax over valid 49 cols; zero the padding -----
        for (int r = wave; r < SP; r += 8) {
            if (r >= SEQ) {
                for (int c = lane; c < SP; c += 32) sc[r * SP + c] = 0.f;
                continue;
            }
            float v0 = (lane      < SEQ) ? sc[r * SP + lane]      : -INFINITY;
            float v1 = (lane + 32 < SEQ) ? sc[r * SP + lane + 32] : -INFINITY;
            float m = fmaxf(v0, v1);
            #pragma unroll
            for (int off = 16; off > 0; off >>= 1) m = fmaxf(m, __shfl_xor(m, off, 32));
            float e0 = (lane      < SEQ) ? expf(v0 - m) : 0.f;
            float e1 = (lane + 32 < SEQ) ? expf(v1 - m) : 0.f;
            float s = e0 + e1;
            #pragma unroll
            for (int off = 16; off > 0; off >>= 1) s += __shfl_xor(s, off, 32);
            const float inv = 1.f / s;
            sc[r * SP + lane]      = e0 * inv;
            sc[r * SP + lane + 32] = e1 * inv;   // cols >= 49 become 0
        }
        __syncthreads();

        // ------- Phase 4: out = attn(64x64) @ v(64x32); write (b,s,h,d) ---
        {
            const int mt = wave >> 1, nt = wave & 1;
            const int row0 = mt * 16, col0 = nt * 16;
            v8f acc = {};
            #pragma unroll
            for (int k0 = 0; k0 < SP; k0 += 4) {
                v2f a, bb;
                a.x  = sc[(row0 + laneM) * SP + k0 + kOff];
                a.y  = sc[(row0 + laneM) * SP + k0 + kOff + 1];
                bb.x = vs[(k0 + kOff)     * HD + col0 + laneM];
                bb.y = vs[(k0 + kOff + 1) * HD + col0 + laneM];
                acc = __builtin_amdgcn_wmma_f32_16x16x4_f32(
                          false, a, false, bb, (short)0, acc, false, false);
            }
            #pragma unroll
            for (int r = 0; r < 8; ++r) {
                const int row = row0 + r + laneHalf * 8;
                const int col = col0 + laneM;
                if (row < SEQ)
                    attn_out[((size_t)b * SEQ + row) * CH + h * HD + col] = acc[r];
            }
        }
        __syncthreads();   // protect qs/ks/vs/sc before next head
    }
}

// ---------------------------------------------------------------------------
// Kernel 2: Y = gelu(X @ W + bias), X:(M,K) W:(K,N) row-major. M%128==0,
// N%16==0, K%64==0. Block = 256 threads (8 waves), tile 128(M) x 16(N),
// X chunk 128x64 staged into LDS with async global->LDS b128 (ASYNCcnt).
// ---------------------------------------------------------------------------
__global__ __launch_bounds__(256) void gemm_gelu(
    const float* __restrict__ X, const float* __restrict__ W,
    const float* __restrict__ bias, float* __restrict__ Y,
    int M, int N, int K)
{
    __shared__ float xs[128 * 68];   // stride 68 floats (272B, 16B aligned)

    const int tid      = threadIdx.x;
    const int lane     = tid & 31;
    const int wave     = tid >> 5;
    const int laneM    = lane & 15;
    const int laneHalf = lane >> 4;
    const int kOff     = laneHalf * 2;
    const int row0     = blockIdx.x * 128;
    const int col0     = blockIdx.y * 16;

    const unsigned xsbase = (unsigned)(uintptr_t)(void*)xs;

    v8f acc = {};
    for (int kc = 0; kc < K; kc += 64) {
        // async stage 128x64 X chunk: 2048 float4 transfers / 256 threads
        for (int i = tid; i < 128 * 16; i += 256) {
            const int r = i >> 4, c4 = i & 15;
            const unsigned ldsa = xsbase + (unsigned)((r * 68 + c4 * 4) * 4);
            const float* gp = &X[(size_t)(row0 + r) * K + kc + c4 * 4];
            asm volatile("global_load_async_to_lds_b128 %0, %1, off"
                         :: "v"(ldsa), "v"(gp) : "memory");
        }
        asm volatile("s_wait_asynccnt 0x0" ::: "memory");
        __syncthreads();

        const int wr = wave * 16;
        #pragma unroll
        for (int kk = 0; kk < 64; kk += 4) {
            v2f a, bb;
            a.x  = xs[(wr + laneM) * 68 + kk + kOff];
            a.y  = xs[(wr + laneM) * 68 + kk + kOff + 1];
            bb.x = W[(size_t)(kc + kk + kOff)     * N + col0 + laneM];  // L2-hot
            bb.y = W[(size_t)(kc + kk + kOff + 1) * N + col0 + laneM];
            acc = __builtin_amdgcn_wmma_f32_16x16x4_f32(
                      false, a, false, bb, (short)0, acc, false, false);
        }
        __syncthreads();
    }

    const float bv = bias[col0 + laneM];
    #pragma unroll
    for (int r = 0; r < 8; ++r) {
        const int row = row0 + wave * 16 + r + laneHalf * 8;
        const int col = col0 + laneM;
        Y[(size_t)row * N + col] = gelu_exact(acc[r] + bv);
    }
}

// ---------------------------------------------------------------------------
extern "C" void kernel_launch(void* const* d_in, const int* in_sizes, int n_in,
                              void* d_out, int out_size, void* d_ws, size_t ws_size,
                              hipStream_t stream) {
    const float* x          = (const float*)d_in[0];
    const float* w_qkv      = (const float*)d_in[1];
    const float* b_qkv      = (const float*)d_in[2];
    const float* bias_table = (const float*)d_in[3];
    const float* w_out      = (const float*)d_in[4];
    const float* b_out      = (const float*)d_in[5];
    float* out = (float*)d_out;
    float* tmp = (float*)d_ws;   // (M, C) fp32 = 308MB scratch

    // 1) fused qkv+attention; attention output parked in d_out
    fused_qkv_attn<<<NB, 256, 0, stream>>>(x, w_qkv, b_qkv, bias_table, out);

    // 2) two chained gelu(X @ w_out + b_out) passes (same weights twice)
    dim3 grid(MTOT / 128, CH / 16);
    gemm_gelu<<<grid, 256, 0, stream>>>(out, w_out, b_out, tmp, MTOT, CH, CH);
    gemm_gelu<<<grid, 256, 0, stream>>>(tmp, w_out, b_out, out, MTOT, CH, CH);
}